// DCT2_1133871366828
// MI455X (gfx1250) — compile-verified
//
#include <hip/hip_runtime.h>
#include <math.h>

// ---------------------------------------------------------------------------
// 2D DCT-II (8 x 2048 x 2048, fp32) as two dense GEMMs per batch using the
// CDNA5 fp32 matrix pipe (V_WMMA_F32_16X16X4_F32).
//   Y1 = X @ C^T   (DCT along last axis)
//   Y2 = C @ Y1    (DCT along second-to-last axis)
// C[k][n] = cos(pi*(2n+1)*k/(2N)), N = 2048 (unnormalized DCT-II).
//
// MI455X reasoning: total ~0.275 TFLOP vs ~268MB of compulsory HBM traffic
// (11.5us @ 23.3 TB/s); DCT matrices (32MB) + per-batch intermediate (16MB)
// stay resident in the 192MB L2, so this is bandwidth-bound and we keep full
// fp32 precision via the native fp32 WMMA rather than downcasting.
// ---------------------------------------------------------------------------

#define NDIM 2048
#define BM   128
#define BN   128
#define BK   32

typedef __attribute__((ext_vector_type(2))) float v2f;
typedef __attribute__((ext_vector_type(8))) float v8f;

// Build the DCT-II matrix and its transpose. Phase is reduced mod 4N (power
// of two) so cosf operates on [0, 2*pi) with full precision.
__global__ __launch_bounds__(256) void gen_dct_mats(float* __restrict__ Cm,
                                                    float* __restrict__ CT) {
    const size_t idx = (size_t)blockIdx.x * blockDim.x + threadIdx.x;
    const int k = (int)(idx / NDIM);
    const int n = (int)(idx % NDIM);
    const unsigned phase = ((2u * (unsigned)n + 1u) * (unsigned)k) & (4u * NDIM - 1u);
    const float ang = 3.14159265358979323846f * (float)phase / (float)(2 * NDIM);
    const float v = cosf(ang);
    Cm[(size_t)k * NDIM + n] = v;
    CT[(size_t)n * NDIM + k] = v;
}

// Row-major NN GEMM: D[2048x2048] = A[2048x2048] * B[2048x2048].
// 256 threads = 8 wave32 waves; block tile 128x128, K-step 32.
// Wave grid 4x2 -> each wave computes a 32x64 macro-tile = 2x4 WMMA tiles.
__global__ __launch_bounds__(256) void dct_gemm_f32(const float* __restrict__ A,
                                                    const float* __restrict__ Bm,
                                                    float* __restrict__ D) {
    __shared__ float sA[BM][BK + 4];   // +4 pad: conflict-free column reads
    __shared__ float sB[BK][BN + 4];

    const int tid    = threadIdx.x;
    const int wave   = tid >> 5;       // 0..7 (wave32)
    const int lane   = tid & 31;
    const int half   = lane >> 4;      // K-half select per ISA A/B layout
    const int lid    = lane & 15;      // M (for A) / N (for B) index
    const int wave_m = wave >> 1;      // 0..3 -> 32-row strips
    const int wave_n = wave & 1;       // 0..1 -> 64-col strips

    const int m0 = blockIdx.y * BM;
    const int n0 = blockIdx.x * BN;

    v8f acc[2][4];
#pragma unroll
    for (int mi = 0; mi < 2; ++mi)
#pragma unroll
        for (int ni = 0; ni < 4; ++ni)
#pragma unroll
            for (int r = 0; r < 8; ++r) acc[mi][ni][r] = 0.0f;

    for (int k0 = 0; k0 < NDIM; k0 += BK) {
        // Cooperative loads: A tile 128x32 and B tile 32x128, as float4.
#pragma unroll
        for (int i = 0; i < 4; ++i) {
            const int idx = i * 256 + tid;
            const int r  = idx >> 3;   // 8 float4 per A row
            const int c4 = idx & 7;
            const float4 v = *(const float4*)(A + (size_t)(m0 + r) * NDIM + k0 + c4 * 4);
            *(float4*)(&sA[r][c4 * 4]) = v;
        }
#pragma unroll
        for (int i = 0; i < 4; ++i) {
            const int idx = i * 256 + tid;
            const int r  = idx >> 5;   // 32 float4 per B row
            const int c4 = idx & 31;
            const float4 v = *(const float4*)(Bm + (size_t)(k0 + r) * NDIM + n0 + c4 * 4);
            *(float4*)(&sB[r][c4 * 4]) = v;
        }
        __syncthreads();

        // 8 K-steps of 4 -> 8x8 = 64 v_wmma_f32_16x16x4_f32 per LDS stage.
#pragma unroll
        for (int kk = 0; kk < BK; kk += 4) {
            // A frag (16x4): lane holds M=lid, VGPR pair covers K = 2*half, 2*half+1.
            v2f afrag[2];
#pragma unroll
            for (int mi = 0; mi < 2; ++mi) {
                const int row = wave_m * 32 + mi * 16 + lid;
                afrag[mi][0] = sA[row][kk + 2 * half];
                afrag[mi][1] = sA[row][kk + 2 * half + 1];
            }
            // B frag (4x16): lane holds N=lid, VGPR pair covers K = 2*half, 2*half+1.
            v2f bfrag[4];
#pragma unroll
            for (int ni = 0; ni < 4; ++ni) {
                const int col = wave_n * 64 + ni * 16 + lid;
                bfrag[ni][0] = sB[kk + 2 * half][col];
                bfrag[ni][1] = sB[kk + 2 * half + 1][col];
            }
#pragma unroll
            for (int mi = 0; mi < 2; ++mi)
#pragma unroll
                for (int ni = 0; ni < 4; ++ni)
                    acc[mi][ni] = __builtin_amdgcn_wmma_f32_16x16x4_f32(
                        /*neg_a=*/false, afrag[mi],
                        /*neg_b=*/false, bfrag[ni],
                        /*c_mod=*/(short)0, acc[mi][ni],
                        /*reuse_a=*/false, /*reuse_b=*/false);
        }
        __syncthreads();
    }

    // C/D layout: VGPR r, lanes 0-15 -> M=r, lanes 16-31 -> M=r+8; N = lid.
#pragma unroll
    for (int mi = 0; mi < 2; ++mi)
#pragma unroll
        for (int ni = 0; ni < 4; ++ni) {
            const int gc = n0 + wave_n * 64 + ni * 16 + lid;
#pragma unroll
            for (int r = 0; r < 8; ++r) {
                const int gr = m0 + wave_m * 32 + mi * 16 + half * 8 + r;
                D[(size_t)gr * NDIM + gc] = acc[mi][ni][r];
            }
        }
}

extern "C" void kernel_launch(void* const* d_in, const int* in_sizes, int n_in,
                              void* d_out, int out_size, void* d_ws, size_t ws_size,
                              hipStream_t stream) {
    (void)n_in; (void)out_size; (void)ws_size;
    const float* x  = (const float*)d_in[0];
    float* out      = (float*)d_out;
    const size_t slice = (size_t)NDIM * NDIM;
    const int batches  = (int)((size_t)in_sizes[0] / slice);   // 8

    // Workspace layout: [C (16MB) | C^T (16MB) | tmp (16MB)] = 48MB.
    float* Cm  = (float*)d_ws;
    float* CT  = Cm + slice;
    float* tmp = CT + slice;

    gen_dct_mats<<<(unsigned)(slice / 256), 256, 0, stream>>>(Cm, CT);

    const dim3 grid(NDIM / BN, NDIM / BM);
    const dim3 block(256);
    for (int b = 0; b < batches; ++b) {
        // Stage 1: Y1 = X_b @ C^T  (DCT along last axis)
        dct_gemm_f32<<<grid, block, 0, stream>>>(x + (size_t)b * slice, CT, tmp);
        // Stage 2: Y2 = C @ Y1     (DCT along rows)
        dct_gemm_f32<<<grid, block, 0, stream>>>(Cm, tmp, out + (size_t)b * slice);
    }
}